// LinkPredictionModel_12128987644486
// MI455X (gfx1250) — compile-verified
//
#include <hip/hip_runtime.h>
#include <hip/hip_bf16.h>

typedef __attribute__((ext_vector_type(2))) float v2f;
typedef __attribute__((ext_vector_type(8))) float v8f;

// ---------------------------------------------------------------------------
// Zero-fill in float4 granularity (b128 stores).
// ---------------------------------------------------------------------------
__global__ void fill_zero_f32v4(float4* __restrict__ p, int n4) {
    int i = blockIdx.x * blockDim.x + threadIdx.x;
    if (i < n4) p[i] = make_float4(0.f, 0.f, 0.f, 0.f);
}

// ---------------------------------------------------------------------------
// Degree count: deg[dst[e]] += 1.0  (float atomics; deg pre-zeroed)
// ---------------------------------------------------------------------------
__global__ void degree_count(const int* __restrict__ dst, float* __restrict__ deg, int e_cnt) {
    int e = blockIdx.x * blockDim.x + threadIdx.x;
    if (e < e_cnt) atomicAdd(&deg[dst[e]], 1.0f);
}

// deg -> dis = rsqrt(deg + 1)  (self-loop included)
__global__ void deg_to_dis(float* __restrict__ dis, int n) {
    int i = blockIdx.x * blockDim.x + threadIdx.x;
    if (i < n) dis[i] = rsqrtf(dis[i] + 1.0f);
}

// Per-edge symmetric norm: norm[e] = dis[src[e]] * dis[dst[e]]
__global__ void edge_norm(const int* __restrict__ src, const int* __restrict__ dst,
                          const float* __restrict__ dis, float* __restrict__ norm, int e_cnt) {
    int e = blockIdx.x * blockDim.x + threadIdx.x;
    if (e < e_cnt) norm[e] = dis[src[e]] * dis[dst[e]];
}

// ---------------------------------------------------------------------------
// H[nrows x 64] = X[nrows x 64] @ W[64 x 64], fp32 WMMA 16x16x4.
// One wave per 16-row tile; 4 N-tiles x 16 K-steps = 64 v_wmma per wave.
//
// Fragment layouts per CDNA5 ISA 7.12.2 (wave32):
//  A (16x4 f32): lanes 0-15 -> M=lane, VGPR{0,1} = K{0,1}; lanes 16-31 -> K{2,3}
//  B (4x16 f32): VGPR0 = rows K0 (lanes 0-15) / K2 (lanes 16-31); VGPR1 = K1/K3
//  C/D (16x16 f32): VGPR v -> row v (lanes 0-15) / row v+8 (lanes 16-31)
// ---------------------------------------------------------------------------
__global__ void gemm64_wmma_f32(const float* __restrict__ X,
                                const float* __restrict__ W,
                                float* __restrict__ H, int nrows) {
    const int wave = (blockIdx.x * blockDim.x + threadIdx.x) >> 5;
    const int lane = threadIdx.x & 31;
    const int m0   = wave * 16;
    if (m0 >= nrows) return;                // whole-wave exit: EXEC stays all-1 for WMMA

    const int half = lane >> 4;             // 0: lanes 0-15, 1: lanes 16-31
    const int l    = lane & 15;

    v8f acc[4] = { (v8f)0.0f, (v8f)0.0f, (v8f)0.0f, (v8f)0.0f };

    const float* xrow = X + (size_t)(m0 + l) * 64;

#pragma unroll
    for (int kt = 0; kt < 16; ++kt) {
        const int k0 = kt * 4 + half * 2;   // this lane-half's K pair
        v2f a;
        a.x = xrow[k0 + 0];
        a.y = xrow[k0 + 1];
#pragma unroll
        for (int nt = 0; nt < 4; ++nt) {
            const int n = nt * 16 + l;
            v2f b;
            b.x = W[(size_t)(k0 + 0) * 64 + n];
            b.y = W[(size_t)(k0 + 1) * 64 + n];
            acc[nt] = __builtin_amdgcn_wmma_f32_16x16x4_f32(
                /*neg_a=*/false, a, /*neg_b=*/false, b,
                /*c_mod=*/(short)0, acc[nt],
                /*reuse_a=*/false, /*reuse_b=*/false);
        }
    }

#pragma unroll
    for (int nt = 0; nt < 4; ++nt) {
#pragma unroll
        for (int v = 0; v < 8; ++v) {
            const int row = m0 + v + half * 8;
            const int col = nt * 16 + l;
            H[(size_t)row * 64 + col] = acc[nt][v];
        }
    }
}

// ---------------------------------------------------------------------------
// Edge scatter: agg[dst] += h[src] * norm[e]
// One thread per (edge, float4 group): 16 threads/edge, one b128 load each
// (16 x 16B = one contiguous 256B slab -> fully coalesced), then 4
// consecutive-address global_atomic_add_f32 (same cacheline burst).
// ---------------------------------------------------------------------------
__global__ void edge_scatter_v4(const float4* __restrict__ h4,
                                const float* __restrict__ norm,
                                const int* __restrict__ src,
                                const int* __restrict__ dst,
                                float* __restrict__ agg, int e_cnt) {
    int tid = blockIdx.x * blockDim.x + threadIdx.x;   // < E*16 = 16M
    if (tid >= e_cnt * 16) return;
    const int e = tid >> 4;
    const int q = tid & 15;
    const int s = src[e];
    const int t = dst[e];
    const float nm = norm[e];
    const float4 v = h4[(size_t)s * 16 + q];
    float* out = agg + (size_t)t * 64 + q * 4;
    atomicAdd(out + 0, v.x * nm);
    atomicAdd(out + 1, v.y * nm);
    atomicAdd(out + 2, v.z * nm);
    atomicAdd(out + 3, v.w * nm);
}

// ---------------------------------------------------------------------------
// out = [relu]( agg + h * dis^2 + b )   (float4 elementwise; safe in place)
// ---------------------------------------------------------------------------
__global__ void finalize_layer_v4(const float4* __restrict__ agg,
                                  const float4* __restrict__ h,
                                  const float* __restrict__ dis,
                                  const float4* __restrict__ bias,
                                  float4* __restrict__ out,
                                  int do_relu, int total4) {
    int tid = blockIdx.x * blockDim.x + threadIdx.x;   // < N*16
    if (tid >= total4) return;
    const int i = tid >> 4;       // node
    const int q = tid & 15;       // float4 group within 64 features
    const float s  = dis[i];
    const float ss = s * s;
    const float4 a  = agg[tid];
    const float4 hv = h[tid];
    const float4 b  = bias[q];
    float4 r;
    r.x = a.x + hv.x * ss + b.x;
    r.y = a.y + hv.y * ss + b.y;
    r.z = a.z + hv.z * ss + b.z;
    r.w = a.w + hv.w * ss + b.w;
    if (do_relu) {
        r.x = fmaxf(r.x, 0.f); r.y = fmaxf(r.y, 0.f);
        r.z = fmaxf(r.z, 0.f); r.w = fmaxf(r.w, 0.f);
    }
    out[tid] = r;
}

// ---------------------------------------------------------------------------
// Host-side launcher
// ---------------------------------------------------------------------------
extern "C" void kernel_launch(void* const* d_in, const int* in_sizes, int n_in,
                              void* d_out, int out_size, void* d_ws, size_t ws_size,
                              hipStream_t stream) {
    const float* x   = (const float*)d_in[0];
    const int*   ei  = (const int*)d_in[1];
    const float* W1  = (const float*)d_in[2];
    const float* b1  = (const float*)d_in[3];
    const float* W2  = (const float*)d_in[4];
    const float* b2  = (const float*)d_in[5];
    float*       out = (float*)d_out;

    const int N  = in_sizes[0] / 64;   // 100000
    const int E  = in_sizes[1] / 2;    // 1000000
    const int NF = N * 64;             // 6.4M floats
    const int N4 = NF / 4;             // float4 count per feature matrix
    const int ET = E * 16;             // scatter threads (float4 granularity)

    const int* src = ei;               // edge_index[0]
    const int* dst = ei + E;           // edge_index[1]

    // Scratch (floats): dis[N] | norm[E] | h[NF] | agg[NF] | h2[NF]  (~81 MB)
    float* ws   = (float*)d_ws;
    float* dis  = ws;
    float* norm = dis + N;
    float* h    = norm + E;
    float* agg  = h + (size_t)NF;
    float* h2   = agg + (size_t)NF;

    const int BLK = 256;
    const int gN   = (N  + BLK - 1) / BLK;
    const int gE   = (E  + BLK - 1) / BLK;
    const int gN4  = (N4 + BLK - 1) / BLK;
    const int gET  = (ET + BLK - 1) / BLK;
    const int gDis4 = ((N + 3) / 4 + BLK - 1) / BLK;
    const int waves = (N + 15) / 16;                  // one wave per 16-row tile
    const int gGemm = (waves + (BLK / 32) - 1) / (BLK / 32);

    // --- degrees -> dis = rsqrt(deg+1) -> per-edge norm ---
    fill_zero_f32v4<<<gDis4, BLK, 0, stream>>>((float4*)dis, (N + 3) / 4);
    degree_count<<<gE, BLK, 0, stream>>>(dst, dis, E);
    deg_to_dis<<<gN, BLK, 0, stream>>>(dis, N);
    edge_norm<<<gE, BLK, 0, stream>>>(src, dst, dis, norm, E);

    // --- layer 1: h = x @ W1 ; agg = scatter(norm * h[src]) ; h = relu(agg + h*dis^2 + b1)
    gemm64_wmma_f32<<<gGemm, BLK, 0, stream>>>(x, W1, h, N);
    fill_zero_f32v4<<<gN4, BLK, 0, stream>>>((float4*)agg, N4);
    edge_scatter_v4<<<gET, BLK, 0, stream>>>((const float4*)h, norm, src, dst, agg, E);
    finalize_layer_v4<<<gN4, BLK, 0, stream>>>((const float4*)agg, (const float4*)h, dis,
                                               (const float4*)b1, (float4*)h, /*relu=*/1, N4);

    // --- layer 2: h2 = h @ W2 ; d_out = scatter + self + b2 (no relu)
    gemm64_wmma_f32<<<gGemm, BLK, 0, stream>>>(h, W2, h2, N);
    fill_zero_f32v4<<<gN4, BLK, 0, stream>>>((float4*)out, N4);
    edge_scatter_v4<<<gET, BLK, 0, stream>>>((const float4*)h2, norm, src, dst, out, E);
    finalize_layer_v4<<<gN4, BLK, 0, stream>>>((const float4*)out, (const float4*)h2, dis,
                                               (const float4*)b2, (float4*)out, /*relu=*/0, N4);
}